// Attention_69587060129983
// MI455X (gfx1250) — compile-verified
//
#include <hip/hip_runtime.h>
#include <math.h>
#include <stdint.h>

// Problem constants (match reference)
#define Bb 2
#define Ss 2048
#define Ee 1024
#define Hh 16
#define Dd 64          // head dim
#define N3 (3*Ee)      // 3072

typedef __attribute__((ext_vector_type(2))) float v2f;
typedef __attribute__((ext_vector_type(8))) float v8f;

// D = A(16x4 f32) * B(4x16 f32) + C(16x16 f32)
#define WMMA_F32(a, b, c) \
    __builtin_amdgcn_wmma_f32_16x16x4_f32(false, (a), false, (b), (short)0, (c), false, false)

// Async-copy one contiguous 4 KB tile (16 rows x 64 f32) global -> LDS.
// 32 lanes x 16 B = 512 B per op; 8 ops per tile, tracked by ASYNCcnt.
__device__ __forceinline__ void async_tile_4k(unsigned lds_byte, const float* gsrc, int lane) {
#pragma unroll
    for (int i = 0; i < 8; ++i) {
        unsigned la = lds_byte + i * 512 + lane * 16;
        const void* ga = (const char*)gsrc + i * 512 + lane * 16;
        asm volatile("global_load_async_to_lds_b128 %0, %1, off"
                     :: "v"(la), "v"(ga) : "memory");
    }
}

// ---------------------------------------------------------------------------
// Kernel 1: QKV projection.  C(M=B*S, N=3E) = X(M,K=E) @ W(K,N) + bias
// One wave -> 4 vertically-stacked 16x16 tiles (reuses each B-fragment 4x).
// Stores Q,K,V in (B,H,S,D) layout.
// ---------------------------------------------------------------------------
__global__ __launch_bounds__(256) void qkv_gemm(const float* __restrict__ X,
                                                const float* __restrict__ W,
                                                const float* __restrict__ bias,
                                                float* __restrict__ Qo,
                                                float* __restrict__ Ko,
                                                float* __restrict__ Vo) {
    const int NT   = N3 / 16;                       // 192 tiles along N
    const int wave = blockIdx.x * 8 + (threadIdx.x >> 5);
    const int lane = threadIdx.x & 31;
    const int half = lane >> 4;
    const int l16  = lane & 15;

    const int mg = wave / NT;                       // 64 M-groups of 64 rows
    const int nt = wave % NT;
    const int m0 = mg * 64;
    const int nglob = nt * 16 + l16;

    const float bv = bias[nglob];
    v8f c[4];
#pragma unroll
    for (int t = 0; t < 4; ++t)
#pragma unroll
        for (int i = 0; i < 8; ++i) c[t][i] = bv;

    const float* arow[4];
#pragma unroll
    for (int t = 0; t < 4; ++t)
        arow[t] = X + (size_t)(m0 + t * 16 + l16) * Ee;

#pragma unroll 4
    for (int k0 = 0; k0 < Ee; k0 += 4) {
        const int kb = k0 + 2 * half;
        v2f b;
        b.x = W[(size_t)kb * N3 + nglob];
        b.y = W[(size_t)(kb + 1) * N3 + nglob];
#pragma unroll
        for (int t = 0; t < 4; ++t) {
            v2f a = *(const v2f*)(arow[t] + kb);
            c[t] = WMMA_F32(a, b, c[t]);
        }
    }

    // Scatter into Q/K/V with (B,H,S,D) layout
    const int tq  = nglob >> 10;
    const int rem = nglob & 1023;
    const int h   = rem >> 6;
    const int d   = rem & 63;
    float* dst = (tq == 0) ? Qo : ((tq == 1) ? Ko : Vo);
#pragma unroll
    for (int t = 0; t < 4; ++t) {
#pragma unroll
        for (int r = 0; r < 8; ++r) {
            const int m  = m0 + t * 16 + r + 8 * half;
            const int bi = m >> 11;
            const int s  = m & 2047;
            dst[(((size_t)(bi * Hh + h) * Ss + s) * Dd) + d] = c[t][r];
        }
    }
}

// ---------------------------------------------------------------------------
// Kernel 2: flash attention with double-buffered async global->LDS staging.
// 2 waves / block; each wave owns one 16-row query tile of one (b,h).
// Per-wave LDS slab: K[2]x4KB | V[2]x4KB | P 1KB  = 17 KB.
// ---------------------------------------------------------------------------
__global__ __launch_bounds__(64) void attn(const float* __restrict__ Qm,
                                           const float* __restrict__ Km,
                                           const float* __restrict__ Vm,
                                           float* __restrict__ Om) {
    __shared__ __align__(16) float smem[2 * 4352];   // 34,816 B (LDS offset 0)

    const int wslot = threadIdx.x >> 5;
    const int wave  = blockIdx.x * 2 + wslot;
    const int lane  = threadIdx.x & 31;
    const int half  = lane >> 4;
    const int l16   = lane & 15;

    const int bh = wave >> 7;                       // / (S/16 = 128)
    const int qt = wave & 127;
    const int q0 = qt * 16;

    const float* Qb = Qm + (size_t)bh * Ss * Dd;
    const float* Kb = Km + (size_t)bh * Ss * Dd;
    const float* Vb = Vm + (size_t)bh * Ss * Dd;

    const int      wbaseF = wslot * 4352;           // float index of slab
    const unsigned wbaseB = (unsigned)wslot * 17408; // byte offset of slab
    float* sKf = smem + wbaseF;                     // + buf*1024
    float* sVf = smem + wbaseF + 2048;              // + buf*1024
    float* sPf = smem + wbaseF + 4096;

    // Preload Q tile as 16 A-fragments (K = 64 -> 16 chunks of depth 4)
    v2f qa[16];
#pragma unroll
    for (int kc = 0; kc < 16; ++kc)
        qa[kc] = *(const v2f*)(Qb + (size_t)(q0 + l16) * Dd + kc * 4 + 2 * half);

    v8f o0 = {}, o1 = {}, o2 = {}, o3 = {};
    float mrow[8], lrow[8];
#pragma unroll
    for (int r = 0; r < 8; ++r) { mrow[r] = -3.0e38f; lrow[r] = 0.0f; }

    // Prime the pipeline: stage key-block 0 (K then V tiles, 16 async ops)
    async_tile_4k(wbaseB + 0,    Kb, lane);
    async_tile_4k(wbaseB + 8192, Vb, lane);

    for (int kb = 0; kb <= qt; ++kb) {
        const int kbase = kb * 16;
        const int cur   = kb & 1;

        // Prefetch next key block into the other buffer, then wait for current
        if (kb < qt) {
            const int nxt = cur ^ 1;
            async_tile_4k(wbaseB + (unsigned)nxt * 4096,        Kb + (size_t)(kb + 1) * 1024, lane);
            async_tile_4k(wbaseB + 8192 + (unsigned)nxt * 4096, Vb + (size_t)(kb + 1) * 1024, lane);
            asm volatile("s_wait_asynccnt 0x10" ::: "memory");  // current block landed
        } else {
            asm volatile("s_wait_asynccnt 0x0" ::: "memory");
        }

        const float* sK = sKf + cur * 1024;
        const float* sV = sVf + cur * 1024;

        // --- scores S = Q K^T (16x16), depth 64, K-frags from LDS ---------
        v8f sc = {};
#pragma unroll
        for (int kc = 0; kc < 16; ++kc) {
            v2f bf = *(const v2f*)(sK + l16 * Dd + kc * 4 + 2 * half);
            sc = WMMA_F32(qa[kc], bf, sc);
        }

        // --- scale, causal mask, online softmax ---------------------------
        const int ncol = kbase + l16;
        float pr[8], fac[8];
#pragma unroll
        for (int r = 0; r < 8; ++r) {
            float s = sc[r] * 0.125f;               // 1/sqrt(64)
            const int mglob = q0 + r + 8 * half;
            if (ncol > mglob) s = -3.0e38f;
            float mx = s;
            mx = fmaxf(mx, __shfl_xor(mx, 1));
            mx = fmaxf(mx, __shfl_xor(mx, 2));
            mx = fmaxf(mx, __shfl_xor(mx, 4));
            mx = fmaxf(mx, __shfl_xor(mx, 8));
            const float mnew = fmaxf(mrow[r], mx);
            const float p = __expf(s - mnew);
            float sum = p;
            sum += __shfl_xor(sum, 1);
            sum += __shfl_xor(sum, 2);
            sum += __shfl_xor(sum, 4);
            sum += __shfl_xor(sum, 8);
            fac[r]  = __expf(mrow[r] - mnew);
            lrow[r] = lrow[r] * fac[r] + sum;
            mrow[r] = mnew;
            pr[r]   = p;
        }
#pragma unroll
        for (int r = 0; r < 8; ++r) {
            o0[r] *= fac[r]; o1[r] *= fac[r]; o2[r] *= fac[r]; o3[r] *= fac[r];
        }

        // --- C-layout -> A-layout via per-wave LDS slab -------------------
#pragma unroll
        for (int r = 0; r < 8; ++r)
            sPf[(r + 8 * half) * 16 + l16] = pr[r];
        __builtin_amdgcn_wave_barrier();            // keep DS store->load order

        // --- O += P V (16x16 @ 16x64), P and V frags from LDS -------------
        const float* pm = sPf + l16 * 16;
#pragma unroll
        for (int kc = 0; kc < 4; ++kc) {
            v2f pa = *(const v2f*)(pm + kc * 4 + 2 * half);
            const int krow = kc * 4 + 2 * half;     // row within V tile
            v2f vb;
            vb.x = sV[krow * Dd + 0 * 16 + l16];
            vb.y = sV[(krow + 1) * Dd + 0 * 16 + l16];
            o0 = WMMA_F32(pa, vb, o0);
            vb.x = sV[krow * Dd + 1 * 16 + l16];
            vb.y = sV[(krow + 1) * Dd + 1 * 16 + l16];
            o1 = WMMA_F32(pa, vb, o1);
            vb.x = sV[krow * Dd + 2 * 16 + l16];
            vb.y = sV[(krow + 1) * Dd + 2 * 16 + l16];
            o2 = WMMA_F32(pa, vb, o2);
            vb.x = sV[krow * Dd + 3 * 16 + l16];
            vb.y = sV[(krow + 1) * Dd + 3 * 16 + l16];
            o3 = WMMA_F32(pa, vb, o3);
        }
    }

    // --- normalize and merge heads: write (B,S,E) -------------------------
    const int bi = bh >> 4;
    const int h  = bh & 15;
#pragma unroll
    for (int r = 0; r < 8; ++r) {
        const float inv = 1.0f / lrow[r];
        const int m = q0 + r + 8 * half;
        float* dst = Om + ((size_t)(bi * Ss + m)) * Ee + h * Dd;
        dst[0 * 16 + l16] = o0[r] * inv;
        dst[1 * 16 + l16] = o1[r] * inv;
        dst[2 * 16 + l16] = o2[r] * inv;
        dst[3 * 16 + l16] = o3[r] * inv;
    }
}

// ---------------------------------------------------------------------------
// Kernel 3: output projection.  out(M,E) = A(M,E) @ Wp(E,E) + bp
// Same 4x M-blocked wave tiling as qkv_gemm.
// ---------------------------------------------------------------------------
__global__ __launch_bounds__(256) void proj_gemm(const float* __restrict__ A,
                                                 const float* __restrict__ W,
                                                 const float* __restrict__ bias,
                                                 float* __restrict__ out) {
    const int NT   = Ee / 16;                       // 64
    const int wave = blockIdx.x * 8 + (threadIdx.x >> 5);
    const int lane = threadIdx.x & 31;
    const int half = lane >> 4;
    const int l16  = lane & 15;

    const int mg = wave / NT;
    const int nt = wave % NT;
    const int m0 = mg * 64;
    const int nglob = nt * 16 + l16;

    const float bv = bias[nglob];
    v8f c[4];
#pragma unroll
    for (int t = 0; t < 4; ++t)
#pragma unroll
        for (int i = 0; i < 8; ++i) c[t][i] = bv;

    const float* arow[4];
#pragma unroll
    for (int t = 0; t < 4; ++t)
        arow[t] = A + (size_t)(m0 + t * 16 + l16) * Ee;

#pragma unroll 4
    for (int k0 = 0; k0 < Ee; k0 += 4) {
        const int kb = k0 + 2 * half;
        v2f b;
        b.x = W[(size_t)kb * Ee + nglob];
        b.y = W[(size_t)(kb + 1) * Ee + nglob];
#pragma unroll
        for (int t = 0; t < 4; ++t) {
            v2f a = *(const v2f*)(arow[t] + kb);
            c[t] = WMMA_F32(a, b, c[t]);
        }
    }

#pragma unroll
    for (int t = 0; t < 4; ++t)
#pragma unroll
        for (int r = 0; r < 8; ++r) {
            const int m = m0 + t * 16 + r + 8 * half;
            out[(size_t)m * Ee + nglob] = c[t][r];
        }
}

// ---------------------------------------------------------------------------
extern "C" void kernel_launch(void* const* d_in, const int* in_sizes, int n_in,
                              void* d_out, int out_size, void* d_ws, size_t ws_size,
                              hipStream_t stream) {
    const float* X     = (const float*)d_in[0];   // (B,S,E)
    const float* Wqkv  = (const float*)d_in[1];   // (E,3E)
    const float* bqkv  = (const float*)d_in[2];   // (3E)
    const float* Wproj = (const float*)d_in[3];   // (E,E)
    const float* bproj = (const float*)d_in[4];   // (E)

    float* ws = (float*)d_ws;
    const size_t BHSD = (size_t)Bb * Hh * Ss * Dd;   // 4,194,304 floats (16 MB)
    float* Q  = ws;
    float* K  = ws + BHSD;
    float* V  = ws + 2 * BHSD;
    float* Ao = ws + 3 * BHSD;                       // attention output (B,S,E)

    // QKV GEMM: 64 M-groups x 192 N-tiles = 12288 waves / 8 per block
    qkv_gemm<<<1536, 256, 0, stream>>>(X, Wqkv, bqkv, Q, K, V);

    // Attention: B*H*(S/16) = 4096 waves / 2 per block (LDS-limited)
    attn<<<2048, 64, 0, stream>>>(Q, K, V, Ao);

    // Output projection: 64 M-groups x 64 N-tiles = 4096 waves / 8 per block
    proj_gemm<<<512, 256, 0, stream>>>(Ao, Wproj, bproj, (float*)d_out);
}